// QEffLlamaSwiftKVAttention_21526376088336
// MI455X (gfx1250) — compile-verified
//
#include <hip/hip_runtime.h>
#include <hip/hip_bf16.h>
#include <math.h>

// ---------------------------------------------------------------------------
// SwiftKV decode attention for MI455X (gfx1250, wave32).
// Memory-bound: 512MB KV stream dominates (~28us @ 23.3 TB/s). All matrix
// math uses V_WMMA_F32_16X16X4_F32 (fp32 matches reference precision; we are
// nowhere near FLOP limits). KV / weight streams use non-temporal loads so
// the one-shot 512MB stream does not evict L2-resident operands.
// ---------------------------------------------------------------------------

#define B_    16
#define HID_  4096
#define NH_   32
#define KVH_  8
#define G_    4
#define D_    128
#define S_    4096
#define SCALE_ 0.08838834764831845f   /* 1/sqrt(128) */
#define MIN_MASKED_ -10000.0f

typedef float v2f __attribute__((ext_vector_type(2)));
typedef float v8f __attribute__((ext_vector_type(8)));

// D(16x16,f32) = A(16x4,f32) * B(4x16,f32) + C
// A: lane m=lane&15 is the M row; VGPRs hold K = {2*(lane>>4), +1}
// B: lane n=lane&15 is the N col; VGPRs hold K = {2*(lane>>4), +1}
// C/D: VGPR r holds row M = r + 8*(lane>>4), col N = lane&15
__device__ __forceinline__ v8f wmma_f32_16x16x4(v2f a, v2f b, v8f c) {
  return __builtin_amdgcn_wmma_f32_16x16x4_f32(
      /*neg_a=*/false, a, /*neg_b=*/false, b,
      /*c_mod=*/(short)0, c, /*reuse_a=*/false, /*reuse_b=*/false);
}

// ---------------------------------------------------------------------------
// C(16 x 4096) = A(16 x 4096) * W^T, W is (4096 x 4096) row-major.
// C[m][n] = sum_k A[m][k] * W[n][k].
// 32 blocks x 256 threads; each wave owns one 16-wide column tile.
// ---------------------------------------------------------------------------
__global__ void gemm_16xN(const float* __restrict__ Am,
                          const float* __restrict__ Wm,
                          float* __restrict__ Cm) {
  const int lane  = threadIdx.x & 31;
  const int wave  = threadIdx.x >> 5;
  const int n0    = (blockIdx.x * 8 + wave) * 16;
  const int idx15 = lane & 15;
  const int kk    = (lane >> 4) * 2;

  const float* arow = Am + (size_t)idx15 * HID_;          // M row (batch)
  const float* brow = Wm + (size_t)(n0 + idx15) * HID_;   // N row of W

  v8f c = {};
#pragma unroll 4
  for (int k = 0; k < HID_; k += 4) {
    v2f a = *(const v2f*)(arow + k + kk);
    v2f b = __builtin_nontemporal_load((const v2f*)(brow + k + kk));
    c = wmma_f32_16x16x4(a, b, c);
  }

  const int mbase = (lane >> 4) * 8;
#pragma unroll
  for (int r = 0; r < 8; ++r)
    Cm[(size_t)(mbase + r) * HID_ + n0 + idx15] = c[r];
}

// ---------------------------------------------------------------------------
// Attention: one workgroup per (b, kvh). 256 threads = 8 waves.
//   prologue: RoPE(q) for the 4 grouped heads -> LDS
//   pass 1  : scores[4][4096] = q . K^T        (WMMA, K nontemporal stream)
//   softmax : scale -> mask -> max -> exp -> sum (LDS tree reductions)
//   pass 2  : ctx[4][128] = probs . V          (WMMA, V nontemporal stream)
// ---------------------------------------------------------------------------
__global__ void attn_kernel(const float* __restrict__ qws,
                            const int* __restrict__ pos_ids,
                            const float* __restrict__ Kc,
                            const float* __restrict__ Vc,
                            const unsigned char* __restrict__ mask,
                            const int* __restrict__ bidx,
                            float* __restrict__ Pws,
                            float* __restrict__ attn_ws) {
  __shared__ float q_s[G_ * D_];     // RoPE'd queries for this kv-head group
  __shared__ float red[256];         // 4 heads x 64-thread reduction
  __shared__ float invsum[G_];

  const int tid = threadIdx.x;
  const int b   = blockIdx.x / KVH_;
  const int kvh = blockIdx.x % KVH_;
  const int bi  = bidx[b] & (B_ - 1);
  const int pos = pos_ids[b];

  // ---- RoPE prologue -------------------------------------------------------
  for (int idx = tid; idx < G_ * D_; idx += blockDim.x) {
    const int h4 = idx >> 7;            // 0..3
    const int d  = idx & (D_ - 1);
    const int head = kvh * G_ + h4;
    const float x  = qws[(size_t)b * HID_ + head * D_ + d];
    const float xp = qws[(size_t)b * HID_ + head * D_ + (d ^ 64)];
    const float rh = (d < 64) ? -xp : xp;
    // inv_freq = 10000^(-(d&63)/64) = exp(-ln(1e4)/64 * (d&63))
    const float invf = __expf(-0.14391157f * (float)(d & 63));
    const float f = (float)pos * invf;
    float sn, cs;
    __sincosf(f, &sn, &cs);
    q_s[idx] = x * cs + rh * sn;
  }
  __syncthreads();

  const int lane  = tid & 31;
  const int wave  = tid >> 5;
  const int idx15 = lane & 15;
  const int kk    = (lane >> 4) * 2;
  const size_t kvbase = ((size_t)bi * KVH_ + kvh) * (size_t)S_ * D_;
  float* Pb = Pws + (size_t)blockIdx.x * G_ * S_;   // scores scratch [4][S]

  // ---- Pass 1: QK^T --------------------------------------------------------
  // C tile: M = 16 key positions, N = heads (0..3 valid).
  const float* qcol = q_s + (idx15 & 3) * D_;       // B col source (clamped)
  for (int st = wave; st < S_ / 16; st += 8) {
    const int s0 = st * 16;
    const float* krow = Kc + kvbase + (size_t)(s0 + idx15) * D_;
    v8f c = {};
#pragma unroll 4
    for (int d = 0; d < D_; d += 4) {
      v2f a  = __builtin_nontemporal_load((const v2f*)(krow + d + kk));
      v2f bq = *(const v2f*)(qcol + d + kk);
      c = wmma_f32_16x16x4(a, bq, c);
    }
    if (idx15 < 4) {                                 // only real head columns
      const int mb = (lane >> 4) * 8;
#pragma unroll
      for (int r = 0; r < 8; ++r)
        Pb[(size_t)idx15 * S_ + s0 + mb + r] = c[r];
    }
  }
  __syncthreads();

  // ---- Softmax (scale, mask, max, exp, sum) --------------------------------
  const int h = tid >> 6;          // head 0..3
  const int t = tid & 63;
  float* Ph = Pb + (size_t)h * S_;
  const unsigned char* mrow = mask + (size_t)b * S_;

  float mx = -3.0e38f;
  for (int j = t; j < S_; j += 64) {
    float v = Ph[j] * SCALE_;
    if (mrow[j]) v = MIN_MASKED_;
    Ph[j] = v;
    mx = fmaxf(mx, v);
  }
  red[tid] = mx;
  __syncthreads();
  for (int off = 32; off > 0; off >>= 1) {
    if (t < off) red[tid] = fmaxf(red[tid], red[tid + off]);
    __syncthreads();
  }
  mx = red[h * 64];
  __syncthreads();

  float sum = 0.0f;
  for (int j = t; j < S_; j += 64) {
    const float e = __expf(Ph[j] - mx);
    Ph[j] = e;
    sum += e;
  }
  red[tid] = sum;
  __syncthreads();
  for (int off = 32; off > 0; off >>= 1) {
    if (t < off) red[tid] += red[tid + off];
    __syncthreads();
  }
  if (t == 0) invsum[h] = 1.0f / red[h * 64];
  __syncthreads();

  // ---- Pass 2: probs x V ---------------------------------------------------
  // C tile: M = heads (0..3 valid), N = 16 d-columns owned by this wave.
  const int d0 = wave * 16;
  const float* Pm = Pb + (size_t)(idx15 & 3) * S_;  // A row source (clamped)
  v8f ctx = {};
#pragma unroll 2
  for (int s = 0; s < S_; s += 4) {
    v2f a = *(const v2f*)(Pm + s + kk);
    const float* vrow = Vc + kvbase + (size_t)(s + kk) * D_ + d0 + idx15;
    v2f bv;
    bv.x = __builtin_nontemporal_load(vrow);
    bv.y = __builtin_nontemporal_load(vrow + D_);
    ctx = wmma_f32_16x16x4(a, bv, ctx);
  }
  if ((lane >> 4) == 0) {          // rows 0..3 live in lanes 0..15, VGPR 0..3
#pragma unroll
    for (int r = 0; r < 4; ++r)
      attn_ws[(size_t)b * HID_ + (kvh * G_ + r) * D_ + d0 + idx15] =
          ctx[r] * invsum[r];
  }
}

// ---------------------------------------------------------------------------
extern "C" void kernel_launch(void* const* d_in, const int* in_sizes, int n_in,
                              void* d_out, int out_size, void* d_ws, size_t ws_size,
                              hipStream_t stream) {
  (void)in_sizes; (void)n_in; (void)out_size; (void)ws_size;
  const float*         hidden  = (const float*)d_in[0];
  const int*           pos_ids = (const int*)d_in[1];
  const float*         Kc      = (const float*)d_in[2];
  const float*         Vc      = (const float*)d_in[3];
  const unsigned char* amask   = (const unsigned char*)d_in[4];
  const int*           bindex  = (const int*)d_in[5];
  const float*         Wq      = (const float*)d_in[6];
  const float*         Wo      = (const float*)d_in[7];

  float* ws      = (float*)d_ws;
  float* qws     = ws;                         // 16*4096 floats (raw Q)
  float* attnws  = ws + (size_t)B_ * HID_;     // 16*4096 floats (ctx)
  float* Pws     = ws + (size_t)2 * B_ * HID_; // 128*4*4096 floats (scores)

  // 1) Q = hidden @ Wq^T
  gemm_16xN<<<dim3(32), dim3(256), 0, stream>>>(hidden, Wq, qws);
  // 2) RoPE + attention per (b, kv-head)
  attn_kernel<<<dim3(B_ * KVH_), dim3(256), 0, stream>>>(
      qws, pos_ids, Kc, Vc, amask, bindex, Pws, attnws);
  // 3) out = ctx @ Wo^T
  gemm_16xN<<<dim3(32), dim3(256), 0, stream>>>(attnws, Wo, (float*)d_out);
}